// LightDGC_936302870781
// MI455X (gfx1250) — compile-verified
//
#include <hip/hip_runtime.h>
#include <hip/hip_bf16.h>
#include <math.h>

typedef __attribute__((ext_vector_type(2))) float v2f;
typedef __attribute__((ext_vector_type(8))) float v8f;

#define B_SZ 8
#define T_SZ 12
#define N_SZ 2000
#define D_SZ 64
#define TOPK 20
#define ALPHA 0.2f
#define EPSV 1e-8f
#define NTILES (N_SZ / 16)   // 125

// ---------------------------------------------------------------------------
// 1) Temporal mean pool + L2 row normalize: X[B,T,N,D] -> Xn[B,N,D]
//    one 64-thread block per (b,n) row
// ---------------------------------------------------------------------------
__global__ void meanpool_norm_kernel(const float* __restrict__ X,
                                     float* __restrict__ Xn) {
    int row = blockIdx.x;            // 0 .. B*N-1
    int b = row / N_SZ;
    int n = row % N_SZ;
    int d = threadIdx.x;             // 0..63
    const float* base = X + (((size_t)b * T_SZ) * N_SZ + n) * D_SZ + d;
    float s = 0.0f;
#pragma unroll
    for (int t = 0; t < T_SZ; ++t) s += base[(size_t)t * N_SZ * D_SZ];
    float m = s * (1.0f / (float)T_SZ);

    __shared__ float red[64];
    red[d] = m * m;
    __syncthreads();
#pragma unroll
    for (int off = 32; off > 0; off >>= 1) {
        if (d < off) red[d] += red[d + off];
        __syncthreads();
    }
    float inv = 1.0f / fmaxf(sqrtf(red[0]), EPSV);
    Xn[((size_t)b * N_SZ + n) * D_SZ + d] = m * inv;
}

// ---------------------------------------------------------------------------
// 2) M = tanh(ALPHA * (E @ theta)),  E:[N,64] theta:[64,64] -> M:[N,64]
// ---------------------------------------------------------------------------
__global__ void embed_gemm_kernel(const float* __restrict__ E,
                                  const float* __restrict__ theta,
                                  float* __restrict__ M) {
    int idx = blockIdx.x * blockDim.x + threadIdx.x;
    if (idx >= N_SZ * D_SZ) return;
    int n = idx / D_SZ, d = idx % D_SZ;
    const float* e = E + (size_t)n * D_SZ;
    float acc = 0.0f;
#pragma unroll 8
    for (int k = 0; k < D_SZ; ++k) acc = fmaf(e[k], theta[k * D_SZ + d], acc);
    M[idx] = tanhf(ALPHA * acc);
}

// ---------------------------------------------------------------------------
// 3) Gate: G = relu(tanh(ALPHA*(M1@M2^T - M2@M1^T)))
//    One wave per 16x16 tile. Transposed tile computed as second WMMA:
//    (M1_i@M2_j^T)^T == M2_i@M1_j^T  -> same C lane layout, per-lane combine.
// ---------------------------------------------------------------------------
__global__ void gate_wmma_kernel(const float* __restrict__ M1,
                                 const float* __restrict__ M2,
                                 float* __restrict__ G) {
    int i0 = blockIdx.y * 16;
    int j0 = blockIdx.x * 16;
    int lane = threadIdx.x;          // 0..31
    int r  = lane & 15;
    int kk = (lane >> 4) << 1;       // 0 or 2

    v8f cA = {};
    v8f cB = {};
#pragma unroll
    for (int k0 = 0; k0 < D_SZ; k0 += 4) {
        v2f a1 = *(const v2f*)(M1 + (size_t)(i0 + r) * D_SZ + k0 + kk);
        v2f b1 = *(const v2f*)(M2 + (size_t)(j0 + r) * D_SZ + k0 + kk);
        v2f a2 = *(const v2f*)(M2 + (size_t)(i0 + r) * D_SZ + k0 + kk);
        v2f b2 = *(const v2f*)(M1 + (size_t)(j0 + r) * D_SZ + k0 + kk);
        cA = __builtin_amdgcn_wmma_f32_16x16x4_f32(false, a1, false, b1,
                                                   (short)0, cA, false, false);
        cB = __builtin_amdgcn_wmma_f32_16x16x4_f32(false, a2, false, b2,
                                                   (short)0, cB, false, false);
    }
    int mbase = (lane >> 4) << 3;    // 0 or 8
    int ncol  = lane & 15;
#pragma unroll
    for (int v = 0; v < 8; ++v) {
        float g = tanhf(ALPHA * (cA[v] - cB[v]));
        G[(size_t)(i0 + mbase + v) * N_SZ + (j0 + ncol)] = fmaxf(g, 0.0f);
    }
}

// ---------------------------------------------------------------------------
// 4a) sim = Xn_b @ Xn_b^T for one batch; one wave per 16x16 tile, K=64
// ---------------------------------------------------------------------------
__global__ void sim_wmma_kernel(const float* __restrict__ Xnb,
                                float* __restrict__ sim) {
    int i0 = blockIdx.y * 16;
    int j0 = blockIdx.x * 16;
    int lane = threadIdx.x;
    int r  = lane & 15;
    int kk = (lane >> 4) << 1;

    v8f c = {};
#pragma unroll
    for (int k0 = 0; k0 < D_SZ; k0 += 4) {
        v2f a = *(const v2f*)(Xnb + (size_t)(i0 + r) * D_SZ + k0 + kk);
        v2f b = *(const v2f*)(Xnb + (size_t)(j0 + r) * D_SZ + k0 + kk);
        c = __builtin_amdgcn_wmma_f32_16x16x4_f32(false, a, false, b,
                                                  (short)0, c, false, false);
    }
    int mbase = (lane >> 4) << 3;
    int ncol  = lane & 15;
#pragma unroll
    for (int v = 0; v < 8; ++v) {
        sim[(size_t)(i0 + mbase + v) * N_SZ + (j0 + ncol)] = c[v];
    }
}

// ---------------------------------------------------------------------------
// 4b) Per-row top-k (k=20) + scatter gate*val into zeroed output row.
//     256 threads per row; 20 x (argmax over LDS with lowest-index tie-break)
// ---------------------------------------------------------------------------
__global__ void topk_scatter_kernel(const float* __restrict__ sim,
                                    const float* __restrict__ G,
                                    float* __restrict__ outb) {
    int row = blockIdx.x;            // 0..N-1
    int tid = threadIdx.x;           // 0..255

    __shared__ float vals[N_SZ];
    __shared__ float selv[TOPK];
    __shared__ int   seli[TOPK];
    __shared__ float redv[256];
    __shared__ int   redi[256];

    const float* srow = sim + (size_t)row * N_SZ;
    for (int j = tid; j < N_SZ; j += 256) vals[j] = srow[j];
    __syncthreads();

    for (int it = 0; it < TOPK; ++it) {
        float bv = -1e30f;
        int   bi = 0x7fffffff;
        for (int j = tid; j < N_SZ; j += 256) {
            float v = vals[j];
            if (v > bv) { bv = v; bi = j; }   // strict > keeps lowest index
        }
        redv[tid] = bv;
        redi[tid] = bi;
        __syncthreads();
#pragma unroll
        for (int off = 128; off > 0; off >>= 1) {
            if (tid < off) {
                float ov = redv[tid + off];
                int   oi = redi[tid + off];
                if (ov > redv[tid] || (ov == redv[tid] && oi < redi[tid])) {
                    redv[tid] = ov;
                    redi[tid] = oi;
                }
            }
            __syncthreads();
        }
        if (tid == 0) {
            selv[it] = redv[0];
            seli[it] = redi[0];
            vals[redi[0]] = -1e30f;          // mask winner
        }
        __syncthreads();
    }

    float* orow = outb + (size_t)row * N_SZ;
    const float* grow = G + (size_t)row * N_SZ;
    for (int j = tid; j < N_SZ; j += 256) orow[j] = 0.0f;
    __syncthreads();
    if (tid < TOPK) {
        int j = seli[tid];
        orow[j] = grow[j] * selv[tid];
    }
}

// ---------------------------------------------------------------------------
extern "C" void kernel_launch(void* const* d_in, const int* in_sizes, int n_in,
                              void* d_out, int out_size, void* d_ws, size_t ws_size,
                              hipStream_t stream) {
    const float* X   = (const float*)d_in[0];
    const float* E1  = (const float*)d_in[1];
    const float* E2  = (const float*)d_in[2];
    const float* th1 = (const float*)d_in[3];
    const float* th2 = (const float*)d_in[4];
    float* out = (float*)d_out;

    // workspace layout (floats): Xn | M1 | M2 | G | sim  (~37 MB)
    float* Xn  = (float*)d_ws;
    float* M1  = Xn + (size_t)B_SZ * N_SZ * D_SZ;
    float* M2  = M1 + (size_t)N_SZ * D_SZ;
    float* G   = M2 + (size_t)N_SZ * D_SZ;
    float* sim = G  + (size_t)N_SZ * N_SZ;

    // 1) mean pool + normalize
    meanpool_norm_kernel<<<dim3(B_SZ * N_SZ), dim3(64), 0, stream>>>(X, Xn);

    // 2) M1, M2
    int ecnt = N_SZ * D_SZ;
    embed_gemm_kernel<<<dim3((ecnt + 255) / 256), dim3(256), 0, stream>>>(E1, th1, M1);
    embed_gemm_kernel<<<dim3((ecnt + 255) / 256), dim3(256), 0, stream>>>(E2, th2, M2);

    // 3) gate G (WMMA)
    gate_wmma_kernel<<<dim3(NTILES, NTILES), dim3(32), 0, stream>>>(M1, M2, G);

    // 4) per batch: sim GEMM (WMMA) then top-k scatter (stream order serializes)
    for (int b = 0; b < B_SZ; ++b) {
        const float* Xnb = Xn + (size_t)b * N_SZ * D_SZ;
        float* outb = out + (size_t)b * N_SZ * N_SZ;
        sim_wmma_kernel<<<dim3(NTILES, NTILES), dim3(32), 0, stream>>>(Xnb, sim);
        topk_scatter_kernel<<<dim3(N_SZ), dim3(256), 0, stream>>>(sim, G, outb);
    }
}